// MultiScaleSubsequenceExtractor_58841051955466
// MI455X (gfx1250) — compile-verified
//
#include <hip/hip_runtime.h>

// CDNA5 / gfx1250: wave32, WMMA (not MFMA), async global->LDS staging.
typedef __attribute__((ext_vector_type(2))) float v2f;
typedef __attribute__((ext_vector_type(8))) float v8f;

#define L_SEQ 1024
#define D_DIM 128
#define R_ROWS 4040            // 1021 + 1017 + 1009 + 993
#define ROWP 132               // padded LDS row stride (floats): (k+2)-row offset -> distinct banks
#define KMAX 48                // 16 starts + 32 (max window) - 1, padded to 48

#if __has_builtin(__builtin_amdgcn_global_load_async_to_lds_b128)
#define HAVE_ASYNC_LDS 1
typedef int v4i_g __attribute__((vector_size(16)));            // matches builtin pointee
typedef __attribute__((address_space(1))) v4i_g gas_v4i;       // global v4i
typedef __attribute__((address_space(3))) v4i_g las_v4i;       // LDS v4i
#else
#define HAVE_ASYNC_LDS 0
#endif

__device__ __forceinline__ void wait_async_zero() {
#if HAVE_ASYNC_LDS
#if __has_builtin(__builtin_amdgcn_s_wait_asynccnt)
  __builtin_amdgcn_s_wait_asynccnt(0);
#else
  asm volatile("s_wait_asynccnt 0" ::: "memory");
#endif
#endif
}

__global__ __launch_bounds__(128)
void mswin_wmma_kernel(const float* __restrict__ seq,
                       const int*  __restrict__ mask,
                       float* __restrict__ out) {
  __shared__ float smem[KMAX * ROWP];   // staged (then masked) rows, ~25.3 KB
  __shared__ float msk[KMAX];           // mask row as float
  __shared__ float invcnt[4 * 16];      // 1/max(window count,1) per (w, start-in-tile)

  const int tid = threadIdx.x;
  const int l0  = blockIdx.x * 16;      // first start position of this tile
  const int b   = blockIdx.y;

  const int c4 = (tid & 31);            // column quad 0..31
  const int kr = (tid >> 5);            // row-within-4

  // ---- Phase 1: mask slice -> LDS; kick off async row staging ---------
  if (tid < KMAX) {
    const int g = l0 + tid;
    msk[tid] = (g < L_SEQ) ? (float)mask[b * L_SEQ + g] : 0.0f;
  }

#if HAVE_ASYNC_LDS
  // Stage raw sequence rows directly global->LDS (ASYNCcnt path, no VGPR
  // round trip). 128 lanes x b128 = 4 rows per iteration.
#pragma unroll
  for (int k0 = 0; k0 < KMAX; k0 += 4) {
    const int k = k0 + kr;
    const int g = l0 + k;
    if (g < L_SEQ) {
      const float* gp = seq + ((size_t)b * L_SEQ + g) * D_DIM + c4 * 4;
      float* lp = &smem[k * ROWP + c4 * 4];
      __builtin_amdgcn_global_load_async_to_lds_b128((gas_v4i*)gp, (las_v4i*)lp,
                                                     /*offset=*/0, /*cpol=*/0);
    }
  }
  wait_async_zero();
  __syncthreads();

  // ---- Phase 2: apply mask in LDS (and zero out-of-range rows) --------
#pragma unroll
  for (int k0 = 0; k0 < KMAX; k0 += 4) {
    const int k = k0 + kr;
    const int g = l0 + k;
    float4 r = make_float4(0.f, 0.f, 0.f, 0.f);
    if (g < L_SEQ) {
      const float mm = msk[k];
      float4 v = *(float4*)(&smem[k * ROWP + c4 * 4]);
      r.x = v.x * mm; r.y = v.y * mm; r.z = v.z * mm; r.w = v.w * mm;
    }
    *(float4*)(&smem[k * ROWP + c4 * 4]) = r;
  }
#else
  __syncthreads();
  // Fallback: load via VGPRs, mask, store to LDS.
#pragma unroll
  for (int k0 = 0; k0 < KMAX; k0 += 4) {
    const int k = k0 + kr;
    const int g = l0 + k;
    float4 v = make_float4(0.f, 0.f, 0.f, 0.f);
    if (g < L_SEQ) {
      v = *(const float4*)(seq + ((size_t)b * L_SEQ + g) * D_DIM + c4 * 4);
    }
    const float mm = msk[k];
    float4 r;
    r.x = v.x * mm; r.y = v.y * mm; r.z = v.z * mm; r.w = v.w * mm;
    *(float4*)(&smem[k * ROWP + c4 * 4]) = r;
  }
#endif

  // Window counts -> reciprocals (reads msk, ordered by the barriers).
  if (tid < 64) {
    const int w_idx = tid >> 4;         // 0..3 -> w = 4,8,16,32
    const int i     = tid & 15;
    const int w     = 4 << w_idx;
    float s = 0.f;
    for (int k = 0; k < w; ++k) s += msk[i + k];
    invcnt[w_idx * 16 + i] = 1.0f / fmaxf(s, 1.0f);
  }
  __syncthreads();

  // ---- Phase 3: banded fp32 WMMA sliding sums -------------------------
  const int lane = tid & 31;
  const int wv   = tid >> 5;            // wave 0..3
  const int m    = lane & 15;           // row / col within 16
  const int hi   = lane >> 4;           // lane half

  const int OFF[4] = {0, 1021, 2038, 3047};   // output row offsets per window size

#pragma unroll
  for (int t2 = 0; t2 < 2; ++t2) {
    const int n = 2 * wv + t2;          // channel tile 0..7
    const int c = 16 * n + m;           // my channel column

    // Preload all 12 B chunks (K rows 4kb..4kb+3, row-striped layout:
    // VGPR0 = rows {0|2} across lane halves, VGPR1 = rows {1|3}).
    v2f bch[12];
#pragma unroll
    for (int kb = 0; kb < 12; ++kb) {
      const int k = 4 * kb + 2 * hi;
      bch[kb].x = smem[k * ROWP + c];
      bch[kb].y = smem[(k + 1) * ROWP + c];
    }

#pragma unroll
    for (int w_idx = 0; w_idx < 4; ++w_idx) {
      const int w   = 4 << w_idx;
      const int nkb = (16 + w - 1 + 3) >> 2;    // 5,6,8,12 k-chunks

      v8f acc = {};
#pragma unroll
      for (int kb = 0; kb < 12; ++kb) {
        if (kb >= nkb) break;                   // folds after unroll of w_idx
        // Banded A (16x4 f32): A[M,k] = 1 iff M <= k < M+w.
        // Layout: VGPR0 = K {0|2} by lane half, VGPR1 = K {1|3}; M = lane&15.
        const int k0 = 4 * kb + 2 * hi;
        v2f a;
        a.x = (k0     >= m && k0     < m + w) ? 1.0f : 0.0f;
        a.y = (k0 + 1 >= m && k0 + 1 < m + w) ? 1.0f : 0.0f;
        acc = __builtin_amdgcn_wmma_f32_16x16x4_f32(
            /*neg_a=*/false, a, /*neg_b=*/false, bch[kb],
            /*c_mod=*/(short)0, acc, /*reuse_a=*/false, /*reuse_b=*/false);
      }

      // D layout: VGPR r holds (M = r + 8*hi, N = lane&15). Scale & store.
#pragma unroll
      for (int r = 0; r < 8; ++r) {
        const int M = r + 8 * hi;
        const int i = l0 + M;                   // window start position
        if (i + w <= L_SEQ) {
          out[((size_t)b * R_ROWS + OFF[w_idx] + i) * D_DIM + c] =
              acc[r] * invcnt[w_idx * 16 + M];
        }
      }
    }
  }
}

extern "C" void kernel_launch(void* const* d_in, const int* in_sizes, int n_in,
                              void* d_out, int out_size, void* d_ws, size_t ws_size,
                              hipStream_t stream) {
  const float* seq  = (const float*)d_in[0];   // (B, L, D) fp32
  const int*   mask = (const int*)d_in[1];     // (B, L) int32
  float* out = (float*)d_out;                  // (B, 4040, D) fp32

  const int B = in_sizes[1] / L_SEQ;           // 128 for the reference setup
  dim3 grid(L_SEQ / 16, B);                    // (64, 128)
  mswin_wmma_kernel<<<grid, 128, 0, stream>>>(seq, mask, out);
}